// GCNGraphClassifier_20375324852533
// MI455X (gfx1250) — compile-verified
//
#include <hip/hip_runtime.h>

#define N_NODES  20000
#define N_EDGES  200000
#define N_GRAPHS 128
#define IN_DIM   16
#define EDGE_DIM 8
#define HID      64
#define OUT_DIM  10

typedef __attribute__((ext_vector_type(2))) float v2f;
typedef __attribute__((ext_vector_type(8))) float v8f;

// D(16x16 f32) = A(16x4 f32) * B(4x16 f32) + C
// 8-arg form: (neg_a, A, neg_b, B, c_mod, C, reuse_a, reuse_b)
#define WMMA_F32(a, b, c) \
  __builtin_amdgcn_wmma_f32_16x16x4_f32(false, (a), false, (b), (short)0, (c), false, false)

static __device__ __forceinline__ v8f splat8(float b) {
  return (v8f){b, b, b, b, b, b, b, b};
}

// ---------------------------------------------------------------------------
// Fold node_emb into w3/b3/root_w (x is a broadcast of a single row).
// w3eff[k,h] = sum_i emb[i]*w3[k, i*HID+h]; rvec[h] = emb@root_w + conv_b
// ---------------------------------------------------------------------------
__global__ void precompute_kernel(const float* __restrict__ node_emb,
                                  const float* __restrict__ w3,
                                  const float* __restrict__ b3,
                                  const float* __restrict__ root_w,
                                  const float* __restrict__ conv_b,
                                  float* __restrict__ w3eff,
                                  float* __restrict__ b3eff,
                                  float* __restrict__ rvec) {
  __shared__ float emb[IN_DIM];
  if (threadIdx.x < IN_DIM) emb[threadIdx.x] = node_emb[threadIdx.x];
  __syncthreads();
  for (int i = threadIdx.x; i < HID * HID; i += blockDim.x) {
    int k = i >> 6;
    int h = i & 63;
    float s = 0.0f;
#pragma unroll
    for (int ii = 0; ii < IN_DIM; ++ii)
      s += emb[ii] * w3[k * (IN_DIM * HID) + ii * HID + h];
    w3eff[i] = s;
  }
  if (threadIdx.x < HID) {
    int h = threadIdx.x;
    float s = 0.0f, rr = 0.0f;
#pragma unroll
    for (int ii = 0; ii < IN_DIM; ++ii) {
      s  += emb[ii] * b3[ii * HID + h];
      rr += emb[ii] * root_w[ii * HID + h];
    }
    b3eff[h] = s;
    rvec[h]  = rr + conv_b[h];
  }
}

// ---------------------------------------------------------------------------
// Edge pipeline: per 16-edge tile (one wave):
//   h1  = relu(EA[16,8]  @ W1[8,64]  + b1)      (2 WMMAs / n-tile)
//   h2  = relu(h1[16,64] @ W2[64,64] + b2)      (16 WMMAs / n-tile)
//   msg =      h2[16,64] @ W3e[64,64]+ b3e      (16 WMMAs / n-tile)
//   agg[dst[e], :] += msg[e, :]                 (f32 global atomics)
// Weights are staged column-major in LDS so B fragments are float2 loads.
// h1/h2 round-trip through a per-wave LDS tile to re-fragment D -> A layout.
// ---------------------------------------------------------------------------
#define NW   4        // waves per block (128 threads)
#define TILE 16
#define HPAD 68       // row pad to dodge LDS bank conflicts

__global__ __launch_bounds__(128) void edge_msg_kernel(
    const float* __restrict__ ea, const int* __restrict__ dstIdx,
    const float* __restrict__ w1, const float* __restrict__ b1,
    const float* __restrict__ w2, const float* __restrict__ b2,
    const float* __restrict__ w3e, const float* __restrict__ b3e,
    float* __restrict__ agg, int nTiles) {
  __shared__ float sW1c[HID * EDGE_DIM];  // [n][k], k fastest
  __shared__ float sW2c[HID * HID];       // [n][k]
  __shared__ float sW3c[HID * HID];       // [n][k]
  __shared__ float sB1[HID], sB2[HID], sB3[HID];
  __shared__ float sH[NW][TILE * HPAD];   // per-wave staging tile [m][k]
  __shared__ int   sDst[NW][TILE];

  const int tid = threadIdx.x;
  for (int i = tid; i < HID * EDGE_DIM; i += blockDim.x) {
    int k = i % EDGE_DIM, n = i / EDGE_DIM;
    sW1c[i] = w1[k * HID + n];
  }
  for (int i = tid; i < HID * HID; i += blockDim.x) {
    int k = i % HID, n = i / HID;
    sW2c[i] = w2[k * HID + n];
    sW3c[i] = w3e[k * HID + n];
  }
  for (int i = tid; i < HID; i += blockDim.x) {
    sB1[i] = b1[i]; sB2[i] = b2[i]; sB3[i] = b3e[i];
  }
  __syncthreads();

  const int lane  = tid & 31;
  const int wave  = tid >> 5;
  const int mrow  = lane & 15;     // A: M index / B,C,D: N index within tile
  const int khalf = lane >> 4;     // K (A/B) or M (C/D) half selector
  float* myH  = sH[wave];
  int*   myDst = sDst[wave];

  const int waveGlobal = blockIdx.x * NW + wave;
  const int waveStride = gridDim.x * NW;

  for (int tile = waveGlobal; tile < nTiles; tile += waveStride) {
    const int e0 = tile * TILE;
    if (lane < TILE) myDst[lane] = dstIdx[e0 + lane];

    // ---- stage 1: h1 = relu(EA @ W1 + b1) -> staging ----
    v2f ea0 = *(const v2f*)&ea[(size_t)(e0 + mrow) * EDGE_DIM + 2 * khalf];
    v2f ea1 = *(const v2f*)&ea[(size_t)(e0 + mrow) * EDGE_DIM + 4 + 2 * khalf];
#pragma unroll
    for (int nt = 0; nt < 4; ++nt) {
      const int n = nt * 16 + mrow;
      v8f acc = splat8(sB1[n]);
      v2f bf0 = *(const v2f*)&sW1c[n * EDGE_DIM + 2 * khalf];
      v2f bf1 = *(const v2f*)&sW1c[n * EDGE_DIM + 4 + 2 * khalf];
      acc = WMMA_F32(ea0, bf0, acc);
      acc = WMMA_F32(ea1, bf1, acc);
#pragma unroll
      for (int j = 0; j < 8; ++j) {
        float v = acc[j];
        myH[(8 * khalf + j) * HPAD + n] = v > 0.0f ? v : 0.0f;
      }
    }

    // ---- stage 2: h2 = relu(h1 @ W2 + b2) -> staging ----
    v2f af[16];
#pragma unroll
    for (int f = 0; f < 16; ++f)
      af[f] = *(const v2f*)&myH[mrow * HPAD + 4 * f + 2 * khalf];
    v8f hacc[4];
#pragma unroll
    for (int nt = 0; nt < 4; ++nt) {
      const int n = nt * 16 + mrow;
      v8f acc = splat8(sB2[n]);
#pragma unroll
      for (int f = 0; f < 16; ++f) {
        v2f bf = *(const v2f*)&sW2c[n * HID + 4 * f + 2 * khalf];
        acc = WMMA_F32(af[f], bf, acc);
      }
      hacc[nt] = acc;
    }
#pragma unroll
    for (int nt = 0; nt < 4; ++nt) {
      const int n = nt * 16 + mrow;
#pragma unroll
      for (int j = 0; j < 8; ++j) {
        float v = hacc[nt][j];
        myH[(8 * khalf + j) * HPAD + n] = v > 0.0f ? v : 0.0f;
      }
    }

    // ---- stage 3: msg = h2 @ W3eff + b3eff, scatter-add to agg ----
#pragma unroll
    for (int f = 0; f < 16; ++f)
      af[f] = *(const v2f*)&myH[mrow * HPAD + 4 * f + 2 * khalf];
#pragma unroll
    for (int nt = 0; nt < 4; ++nt) {
      const int n = nt * 16 + mrow;
      v8f acc = splat8(sB3[n]);
#pragma unroll
      for (int f = 0; f < 16; ++f) {
        v2f bf = *(const v2f*)&sW3c[n * HID + 4 * f + 2 * khalf];
        acc = WMMA_F32(af[f], bf, acc);
      }
#pragma unroll
      for (int j = 0; j < 8; ++j) {
        const int m = 8 * khalf + j;
        unsafeAtomicAdd(&agg[(size_t)myDst[m] * HID + n], acc[j]);
      }
    }
  }
}

// ---------------------------------------------------------------------------
// out = relu(agg + r); segment-sum into per-graph sums + counts
// ---------------------------------------------------------------------------
__global__ void node_pool_kernel(const float* __restrict__ agg,
                                 const float* __restrict__ rvec,
                                 const int* __restrict__ batch,
                                 float* __restrict__ sums,
                                 float* __restrict__ counts) {
  int idx = blockIdx.x * blockDim.x + threadIdx.x;
  if (idx >= N_NODES * HID) return;
  int node = idx >> 6;
  int h    = idx & 63;
  float v = agg[idx] + rvec[h];
  v = v > 0.0f ? v : 0.0f;
  int g = batch[node];
  unsafeAtomicAdd(&sums[g * HID + h], v);
  if (h == 0) unsafeAtomicAdd(&counts[g], 1.0f);
}

// ---------------------------------------------------------------------------
// pooled = sums/max(counts,1); logits = relu(pooled@l1+b)@l2+b  (tiny)
// ---------------------------------------------------------------------------
__global__ void head_kernel(const float* __restrict__ sums,
                            const float* __restrict__ counts,
                            const float* __restrict__ l1w,
                            const float* __restrict__ l1b,
                            const float* __restrict__ l2w,
                            const float* __restrict__ l2b,
                            float* __restrict__ out) {
  int g = threadIdx.x;
  if (g >= N_GRAPHS) return;
  float c = counts[g];
  c = c > 1.0f ? c : 1.0f;
  float inv = 1.0f / c;
  float pooled[HID];
#pragma unroll
  for (int h = 0; h < HID; ++h) pooled[h] = sums[g * HID + h] * inv;
  float logit[OUT_DIM];
#pragma unroll
  for (int o = 0; o < OUT_DIM; ++o) logit[o] = l2b[o];
  for (int j = 0; j < 2 * HID; ++j) {
    float t = l1b[j];
#pragma unroll
    for (int h = 0; h < HID; ++h) t += pooled[h] * l1w[h * (2 * HID) + j];
    t = t > 0.0f ? t : 0.0f;
#pragma unroll
    for (int o = 0; o < OUT_DIM; ++o) logit[o] += t * l2w[j * OUT_DIM + o];
  }
#pragma unroll
  for (int o = 0; o < OUT_DIM; ++o) out[g * OUT_DIM + o] = logit[o];
}

extern "C" void kernel_launch(void* const* d_in, const int* in_sizes, int n_in,
                              void* d_out, int out_size, void* d_ws, size_t ws_size,
                              hipStream_t stream) {
  const float* edge_attr = (const float*)d_in[0];
  const int*   edge_index = (const int*)d_in[1];
  const int*   batch = (const int*)d_in[2];
  const float* node_emb = (const float*)d_in[3];
  const float* w1 = (const float*)d_in[4];
  const float* b1 = (const float*)d_in[5];
  const float* w2 = (const float*)d_in[6];
  const float* b2 = (const float*)d_in[7];
  const float* w3 = (const float*)d_in[8];
  const float* b3 = (const float*)d_in[9];
  const float* root_w = (const float*)d_in[10];
  const float* conv_b = (const float*)d_in[11];
  const float* l1w = (const float*)d_in[12];
  const float* l1b = (const float*)d_in[13];
  const float* l2w = (const float*)d_in[14];
  const float* l2b = (const float*)d_in[15];

  float* ws     = (float*)d_ws;
  float* w3eff  = ws;                        // HID*HID
  float* b3eff  = w3eff + HID * HID;         // HID
  float* rvec   = b3eff + HID;               // HID
  float* sums   = rvec + HID;                // N_GRAPHS*HID
  float* counts = sums + N_GRAPHS * HID;     // N_GRAPHS
  float* agg    = counts + N_GRAPHS;         // N_NODES*HID

  size_t zeroFloats = (size_t)N_GRAPHS * HID + N_GRAPHS + (size_t)N_NODES * HID;
  hipMemsetAsync(sums, 0, zeroFloats * sizeof(float), stream);

  precompute_kernel<<<1, 256, 0, stream>>>(node_emb, w3, b3, root_w, conv_b,
                                           w3eff, b3eff, rvec);

  const int nTiles = N_EDGES / TILE;  // 12500, exact
  edge_msg_kernel<<<512, 128, 0, stream>>>(edge_attr, edge_index + N_EDGES,
                                           w1, b1, w2, b2, w3eff, b3eff,
                                           agg, nTiles);

  const int nodeThreads = N_NODES * HID;
  node_pool_kernel<<<(nodeThreads + 255) / 256, 256, 0, stream>>>(
      agg, rvec, batch, sums, counts);

  head_kernel<<<1, 128, 0, stream>>>(sums, counts, l1w, l1b, l2w, l2b,
                                     (float*)d_out);
}